// SAGE_63522566307932
// MI455X (gfx1250) — compile-verified
//
#include <hip/hip_runtime.h>

typedef __attribute__((ext_vector_type(2))) float v2f;
typedef __attribute__((ext_vector_type(8))) float v8f;

#define N_NODES 100000
#define N_EDGES 1600000
#define D_IN    128
#define D_HID   128
#define D_OUT   64

// LDS B-tile stride: 258 floats (pad from 256) -> even (8B-aligned b64 rows),
// and rows land on distinct banks (258 % 64 == 2).
#define BSTRIDE 258

// ---------------------------------------------------------------------------
// Zero a float buffer (grid-stride).
// ---------------------------------------------------------------------------
__global__ void sage_zero_f32(float* __restrict__ p, long n) {
    long i      = (long)blockIdx.x * blockDim.x + threadIdx.x;
    long stride = (long)gridDim.x * blockDim.x;
    for (; i < n; i += stride) p[i] = 0.0f;
}

// ---------------------------------------------------------------------------
// In-degree as float (matches reference segment_sum of ones).
// ---------------------------------------------------------------------------
__global__ void sage_degree(const int* __restrict__ dst, float* __restrict__ deg,
                            int n_edges) {
    int i = blockIdx.x * blockDim.x + threadIdx.x;
    if (i < n_edges) atomicAdd(&deg[dst[i]], 1.0f);
}

// ---------------------------------------------------------------------------
// Edge scatter-add: one wave per edge, lane l moves float4 #l of the 128-f32
// feature row. Gather is a coalesced 512B row read; the atomic RMWs stay in
// L2 (the 51MB accumulator is L2-resident on MI455X's 192MB L2).
// ---------------------------------------------------------------------------
__global__ void sage_scatter(const float* __restrict__ feat,
                             const int* __restrict__ src,
                             const int* __restrict__ dst,
                             float* __restrict__ agg, int n_edges) {
    int wave = (int)((blockIdx.x * (long)blockDim.x + threadIdx.x) >> 5);
    int lane = threadIdx.x & 31;
    if (wave >= n_edges) return;                      // wave-uniform exit
    int s = src[wave];
    int d = dst[wave];
    const float4* fr = (const float4*)(feat + (long)s * D_IN);
    float*        to = agg + (long)d * D_IN;
    float4 v = fr[lane];
    atomicAdd(&to[lane * 4 + 0], v.x);
    atomicAdd(&to[lane * 4 + 1], v.y);
    atomicAdd(&to[lane * 4 + 2], v.z);
    atomicAdd(&to[lane * 4 + 3], v.w);
}

// ---------------------------------------------------------------------------
// Fused SAGE layer GEMM via FP32 WMMA:
//   Out = act( Aself @ Wself + (Aagg/max(deg,1)) @ Wneigh + bias )
//
// Block = 256 threads = 8 waves. All 8 waves share ONE N-tile; wave w owns
// M-tile (mgroup*8 + w). The 256x16 concatenated B tile [Wself;Wneigh] is
// cooperatively staged into LDS once per block (transposed, padded stride),
// so each lane's K-pair is a single aligned ds_load_b64 with an immediate
// offset, and each A K-pair is a global_load_b64 with an immediate offset.
// K = 256 -> 64 V_WMMA_F32_16X16X4_F32 per tile, fully unrolled, branch-free.
//
// Lane layouts (ISA 7.12.2, wave32):
//   A 16x4 f32 : lane L -> row (L&15); vgprs hold K = 2*(L>>4), +1
//   B 4x16 f32 : lane L -> col (L&15); vgprs hold K = 2*(L>>4), +1
//   C/D 16x16  : vgpr r -> row r + 8*(L>>4), col (L&15)
// ---------------------------------------------------------------------------
template <int NCOLS, bool RELU>
__global__ __launch_bounds__(256)
void sage_gemm_wmma(const float* __restrict__ Aself,   // [n_rows, 128]
                    const float* __restrict__ Aagg,    // [n_rows, 128] (sums)
                    const float* __restrict__ deg,     // [n_rows]
                    const float* __restrict__ Wself,   // [128, NCOLS]
                    const float* __restrict__ Wneigh,  // [128, NCOLS]
                    const float* __restrict__ bias,    // [NCOLS]
                    float* __restrict__ Out,           // [n_rows, NCOLS]
                    int n_rows) {
    constexpr int NTN = NCOLS / 16;
    __shared__ float Bs[16 * BSTRIDE];                 // 16 cols x 256(+2) K

    const int ntile  = blockIdx.x % NTN;
    const int mgroup = blockIdx.x / NTN;
    const int wave   = threadIdx.x >> 5;
    const int mtile  = mgroup * 8 + wave;

    // --- cooperative B stage: thread t owns K-row t (t = 0..255) ----------
    {
        const int t = threadIdx.x;
        const float* wrow = (t < D_IN)
                                ? (Wself  + (size_t)t          * NCOLS)
                                : (Wneigh + (size_t)(t - D_IN) * NCOLS);
        const float4* srcv = (const float4*)(wrow + ntile * 16);
        #pragma unroll
        for (int c4 = 0; c4 < 4; ++c4) {
            float4 v = srcv[c4];
            Bs[(c4 * 4 + 0) * BSTRIDE + t] = v.x;
            Bs[(c4 * 4 + 1) * BSTRIDE + t] = v.y;
            Bs[(c4 * 4 + 2) * BSTRIDE + t] = v.z;
            Bs[(c4 * 4 + 3) * BSTRIDE + t] = v.w;
        }
    }
    __syncthreads();

    if (mtile * 16 >= n_rows) return;                  // wave-uniform exit

    const int lane = threadIdx.x & 31;
    const int half = lane >> 4;                        // 0 | 1
    const int l16  = lane & 15;
    const int koff = 2 * half;                         // this lane's K phase
    const int arow = mtile * 16 + l16;
    const int bcol = ntile * 16 + l16;

    float r = deg[arow];
    r = 1.0f / fmaxf(r, 1.0f);                         // mean normalization

    const float* aP = Aself + (size_t)arow * D_IN + koff;
    const float* gP = Aagg  + (size_t)arow * D_IN + koff;
    const float* bP = &Bs[l16 * BSTRIDE + koff];       // LDS, immediate offsets

    v8f acc = {};

    // --- self half: K = 0..127 --------------------------------------------
    #pragma unroll
    for (int k0 = 0; k0 < D_IN; k0 += 4) {
        v2f a = *(const v2f*)(aP + k0);
        v2f b = *(const v2f*)(bP + k0);
        acc = __builtin_amdgcn_wmma_f32_16x16x4_f32(
            false, a, false, b, (short)0, acc, false, false);
    }

    // --- neighbor half: K = 128..255 (A scaled by 1/deg) ------------------
    const float* bP2 = bP + D_IN;
    #pragma unroll
    for (int k0 = 0; k0 < D_IN; k0 += 4) {
        v2f a = *(const v2f*)(gP + k0);
        a.x *= r;
        a.y *= r;
        v2f b = *(const v2f*)(bP2 + k0);
        acc = __builtin_amdgcn_wmma_f32_16x16x4_f32(
            false, a, false, b, (short)0, acc, false, false);
    }

    // --- epilogue: bias (+ ReLU), C/D layout writeback --------------------
    const float bv = bias[bcol];
    #pragma unroll
    for (int rr = 0; rr < 8; ++rr) {
        int orow = mtile * 16 + rr + 8 * half;
        float v = acc[rr] + bv;
        if (RELU) v = fmaxf(v, 0.0f);
        Out[(size_t)orow * NCOLS + bcol] = v;
    }
}

// ---------------------------------------------------------------------------
// Launch: zero -> degree -> scatter1 -> gemm1(relu) -> zero agg ->
//         scatter2 -> gemm2
// ---------------------------------------------------------------------------
extern "C" void kernel_launch(void* const* d_in, const int* in_sizes, int n_in,
                              void* d_out, int out_size, void* d_ws, size_t ws_size,
                              hipStream_t stream) {
    const float* x        = (const float*)d_in[0];
    const int*   src      = (const int*)  d_in[1];
    const int*   dst      = (const int*)  d_in[2];
    const float* W_self1  = (const float*)d_in[3];
    const float* W_neigh1 = (const float*)d_in[4];
    const float* b1       = (const float*)d_in[5];
    const float* W_self2  = (const float*)d_in[6];
    const float* W_neigh2 = (const float*)d_in[7];
    const float* b2       = (const float*)d_in[8];
    float*       out      = (float*)d_out;

    float* deg = (float*)d_ws;                         // N_NODES
    float* agg = deg + N_NODES;                        // N_NODES * 128
    float* h   = agg + (size_t)N_NODES * D_IN;         // N_NODES * 128

    const int MTILES  = N_NODES / 16;                  // 6250
    const int MGROUPS = (MTILES + 7) / 8;              // 782

    // zero deg + agg (contiguous)
    sage_zero_f32<<<2048, 256, 0, stream>>>(deg, (long)N_NODES + (long)N_NODES * D_IN);

    // float in-degree
    sage_degree<<<(N_EDGES + 255) / 256, 256, 0, stream>>>(dst, deg, N_EDGES);

    // layer 1 aggregation (sum); one wave per edge
    {
        long threads = (long)N_EDGES * 32;
        int blocks = (int)((threads + 255) / 256);
        sage_scatter<<<blocks, 256, 0, stream>>>(x, src, dst, agg, N_EDGES);
    }

    // layer 1 fused GEMM + bias + ReLU  (N=128 -> 8 ntiles)
    sage_gemm_wmma<D_HID, true><<<MGROUPS * (D_HID / 16), 256, 0, stream>>>(
        x, agg, deg, W_self1, W_neigh1, b1, h, N_NODES);

    // re-zero agg for layer 2
    sage_zero_f32<<<2048, 256, 0, stream>>>(agg, (long)N_NODES * D_HID);

    // layer 2 aggregation of h
    {
        long threads = (long)N_EDGES * 32;
        int blocks = (int)((threads + 255) / 256);
        sage_scatter<<<blocks, 256, 0, stream>>>(h, src, dst, agg, N_EDGES);
    }

    // layer 2 fused GEMM + bias (no ReLU)  (N=64 -> 4 ntiles)
    sage_gemm_wmma<D_OUT, false><<<MGROUPS * (D_OUT / 16), 256, 0, stream>>>(
        h, agg, deg, W_self2, W_neigh2, b2, out, N_NODES);
}